// Attention_4277787427740
// MI455X (gfx1250) — compile-verified
//
#include <hip/hip_runtime.h>

// ---- problem constants (match reference) ----
#define BATCH      32
#define NUM_BLOCKS 2048
#define PAGE_TOK   16
#define KVH        8
#define NH         32
#define HD         128
#define MAXB       128
#define GQA        4                      // NH / KVH
#define QK_SCALE   0.08838834764831845f   // 1/sqrt(128)

#define WAVES        4
#define PAGES_PER_WV 32
#define ROWF         132                  // padded floats per LDS K row (128 + 4)

typedef float v2f __attribute__((ext_vector_type(2)));
typedef float v8f __attribute__((ext_vector_type(8)));

__device__ __forceinline__ v8f wmma_f32_16x16x4(v2f a, v2f b, v8f c) {
  return __builtin_amdgcn_wmma_f32_16x16x4_f32(
      /*neg_a=*/false, a, /*neg_b=*/false, b,
      /*c_mod=*/(short)0, c, /*reuse_a=*/false, /*reuse_b=*/false);
}

// CDNA5 async copy: 16B/lane global -> LDS, tracked by ASYNCcnt (no VGPR transit)
__device__ __forceinline__ void async_b128_to_lds(unsigned lds_off, const void* gptr) {
  asm volatile("global_load_async_to_lds_b128 %0, %1, off"
               :: "v"(lds_off), "v"(gptr) : "memory");
}
__device__ __forceinline__ void wait_asynccnt0() {
  asm volatile("s_wait_asynccnt 0" ::: "memory");
}

__global__ __launch_bounds__(128) void pa_decode_kernel(
    const float* __restrict__ q,       // [B,1,NH,HD]
    const float* __restrict__ knew,    // [B,KVH,HD]
    const float* __restrict__ vnew,    // [B,KVH,HD]
    const float* __restrict__ kcache,  // [NB,16,KVH,HD]
    const float* __restrict__ vcache,  // [NB,16,KVH,HD]
    const int*   __restrict__ ctxlen,  // [B]
    const int*   __restrict__ btab,    // [B,MAXB]
    const int*   __restrict__ smap,    // [B] flat slots
    float*       __restrict__ out)     // [B,1,NH,HD]
{
  __shared__ float kb[WAVES][PAGE_TOK * ROWF];   // staged K page per wave
  __shared__ int   ovr[WAVES][PAGE_TOK];         // per-row new-token override idx
  __shared__ float Oar[WAVES][GQA][HD];          // partial outputs
  __shared__ float mar[WAVES][GQA];
  __shared__ float sar[WAVES][GQA];

  const int b    = blockIdx.x >> 3;
  const int kvh  = blockIdx.x & 7;
  const int lane = threadIdx.x & 31;
  const int wv   = threadIdx.x >> 5;
  const int ln16 = lane & 15;           // row-in-tile / q-head column
  const int h    = lane >> 4;           // wave half

  const int sm_lane = smap[lane];       // all 32 slot_mapping entries, one per lane

  // ---- Q as WMMA B-fragment (K=dims x N=qhead), pre-scaled, 32 chunks of K=4 ----
  v2f qb[32];
  {
    const bool valid = ln16 < GQA;
    const float* qbase = q + (size_t)(b * NH + kvh * GQA + (valid ? ln16 : 0)) * HD + 2 * h;
#pragma unroll
    for (int kk = 0; kk < 32; ++kk) {
      float x = 0.f, y = 0.f;
      if (valid) { x = qbase[kk * 4] * QK_SCALE; y = qbase[kk * 4 + 1] * QK_SCALE; }
      qb[kk].x = x; qb[kk].y = y;
    }
  }

  const int ctx    = ctxlen[b];
  const int ptotal = (ctx + PAGE_TOK - 1) >> 4;
  const int ps     = wv * PAGES_PER_WV;
  const int pe     = min(ps + PAGES_PER_WV, ptotal);

  float m = -1e30f, s = 0.f;
  v8f O[8];
  {
    v8f z = {0.f,0.f,0.f,0.f,0.f,0.f,0.f,0.f};
#pragma unroll
    for (int t = 0; t < 8; ++t) O[t] = z;
  }

  float* kbw  = kb[wv];
  int*   ovrw = ovr[wv];
  // low 32 bits of a generic LDS pointer == LDS byte offset (ISA 10.2 aperture map)
  const unsigned kbw_off = (unsigned)(uintptr_t)kbw + (unsigned)(lane * 16);

  for (int p = ps; p < pe; ++p) {
    const int blk = btab[b * MAXB + p];

    // ---- async-stage K page into LDS (16B/lane per row), build override table ----
#pragma unroll 4
    for (int t = 0; t < PAGE_TOK; ++t) {
      const int flat = blk * PAGE_TOK + t;
      unsigned long long mb = __ballot(sm_lane == flat);   // wave-uniform
      const char* ksrc;
      int j = -1;
      if (mb) {
        j = (int)(__ffsll(mb) - 1);
        ksrc = (const char*)(knew + (size_t)(j * KVH + kvh) * HD);
      } else {
        ksrc = (const char*)(kcache + ((size_t)flat * KVH + kvh) * HD);
      }
      if (lane == 0) ovrw[t] = j;
      async_b128_to_lds(kbw_off + (unsigned)(t * ROWF * 4), ksrc + lane * 16);
    }

    // ---- prefetch next page K/V lines (global_prefetch_b8) ----
    if (p + 1 < pe) {
      const int nblk = btab[b * MAXB + p + 1];
#pragma unroll
      for (int it = 0; it < 4; ++it) {
        int idx = it * 32 + lane;                       // (row, 64B chunk)
        int r_ = idx >> 3, ch = idx & 7;
        size_t off = ((size_t)(nblk * PAGE_TOK + r_) * KVH + kvh) * HD + ch * 16;
        __builtin_prefetch(kcache + off, 0, 1);
        __builtin_prefetch(vcache + off, 0, 1);
      }
    }

    wait_asynccnt0();   // K page resident in LDS

    // ---- QK^T: C[tok, qh] accumulated over D=128 in 32 K=4 steps ----
    v8f c = {0.f,0.f,0.f,0.f,0.f,0.f,0.f,0.f};
#pragma unroll
    for (int kk = 0; kk < 32; ++kk) {
      v2f a = *(const v2f*)&kbw[ln16 * ROWF + kk * 4 + 2 * h];  // A: K rows
      c = wmma_f32_16x16x4(a, qb[kk], c);
    }

    // ---- mask invalid tokens in last partial page ----
    const int nv = ctx - p * PAGE_TOK;
    if (nv < PAGE_TOK) {
#pragma unroll
      for (int r = 0; r < 8; ++r) {
        int tok = r + 8 * h;
        if (tok >= nv) c[r] = -1e30f;
      }
    }

    // ---- online softmax update (per qh column = per lane%16) ----
    float pm = c[0];
#pragma unroll
    for (int r = 1; r < 8; ++r) pm = fmaxf(pm, c[r]);
    pm = fmaxf(pm, __shfl_xor(pm, 16, 32));
    const float mn = fmaxf(m, pm);
    const float f  = __expf(m - mn);
    m = mn;
#pragma unroll
    for (int r = 0; r < 8; ++r) c[r] = __expf(c[r] - mn);
    float psum = c[0];
#pragma unroll
    for (int r = 1; r < 8; ++r) psum += c[r];
    psum += __shfl_xor(psum, 16, 32);
    s = s * f + psum;

    // rescale O: O row qh = r + 8h needs f from lane (r + 8h)
    float fr[8];
#pragma unroll
    for (int r = 0; r < 8; ++r) fr[r] = __shfl(f, r + 8 * h, 32);
#pragma unroll
    for (int dt = 0; dt < 8; ++dt)
#pragma unroll
      for (int r = 0; r < 8; ++r) O[dt][r] *= fr[r];

    // ---- P^T x V: 4 token-chunks of K=4, 8 dim tiles of N=16 ----
#pragma unroll
    for (int kc = 0; kc < 4; ++kc) {
      // A-fragment: lane holds row qh=lane%16, tokens kc*4 + 2h + {0,1}
      v2f a;
#pragma unroll
      for (int v = 0; v < 2; ++v) {
        const int t0 = kc * 4 + v;       // token for h==0 lanes
        const int t1 = kc * 4 + 2 + v;   // token for h==1 lanes
        float v0 = (t0 < 8) ? c[t0] : __shfl_xor(c[t0 & 7], 16, 32);
        float v1 = (t1 >= 8) ? c[t1 & 7] : __shfl_xor(c[t1 & 7], 16, 32);
        float av = h ? v1 : v0;
        if (v == 0) a.x = av; else a.y = av;
      }
      // V B-fragment rows (with new-token override), streamed NT from global
      const float* vrow[2];
#pragma unroll
      for (int v = 0; v < 2; ++v) {
        int tok = kc * 4 + 2 * h + v;
        int ov  = ovrw[tok];
        vrow[v] = (ov >= 0)
            ? vnew + (size_t)(ov * KVH + kvh) * HD
            : vcache + ((size_t)(blk * PAGE_TOK + tok) * KVH + kvh) * HD;
      }
#pragma unroll
      for (int dt = 0; dt < 8; ++dt) {
        v2f bv;
        bv.x = __builtin_nontemporal_load(&vrow[0][dt * 16 + ln16]);
        bv.y = __builtin_nontemporal_load(&vrow[1][dt * 16 + ln16]);
        O[dt] = wmma_f32_16x16x4(a, bv, O[dt]);
      }
    }
  }

  // ---- publish per-wave partials ----
  if (lane < GQA) { mar[wv][lane] = m; sar[wv][lane] = s; }
#pragma unroll
  for (int dt = 0; dt < 8; ++dt)
#pragma unroll
    for (int r = 0; r < GQA; ++r)
      if (h == 0) Oar[wv][r][dt * 16 + ln16] = O[dt][r];

  __syncthreads();

  // ---- merge 4 wave partials, normalize, write out ----
  for (int i = threadIdx.x; i < GQA * HD; i += blockDim.x) {
    const int qh2 = i >> 7, d = i & 127;
    float mx = mar[0][qh2];
#pragma unroll
    for (int w = 1; w < WAVES; ++w) mx = fmaxf(mx, mar[w][qh2]);
    float ss = 0.f, vv = 0.f;
#pragma unroll
    for (int w = 0; w < WAVES; ++w) {
      float e = __expf(mar[w][qh2] - mx);
      ss += sar[w][qh2] * e;
      vv += Oar[w][qh2][d] * e;
    }
    out[(size_t)(b * NH + kvh * GQA + qh2) * HD + d] = vv / ss;
  }
}

extern "C" void kernel_launch(void* const* d_in, const int* in_sizes, int n_in,
                              void* d_out, int out_size, void* d_ws, size_t ws_size,
                              hipStream_t stream) {
  (void)in_sizes; (void)n_in; (void)out_size; (void)d_ws; (void)ws_size;
  pa_decode_kernel<<<dim3(BATCH * KVH), dim3(128), 0, stream>>>(
      (const float*)d_in[0], (const float*)d_in[1], (const float*)d_in[2],
      (const float*)d_in[3], (const float*)d_in[4], (const int*)d_in[5],
      (const int*)d_in[6], (const int*)d_in[7], (float*)d_out);
}